// CalculateDirectionFeature_27376121545514
// MI455X (gfx1250) — compile-verified
//
#include <hip/hip_runtime.h>
#include <math.h>

typedef __attribute__((ext_vector_type(2))) float v2f;
typedef __attribute__((ext_vector_type(8))) float v8f;

#define B_    4
#define P_    6
#define F_    257
#define T_    300
#define N_    36
#define NPAD  48    // 3 tiles of 16 rows (n), zero padded past 36
#define TPAD  304   // LDS row stride for the obs-trig table
#define NT_TILES 3
#define TT_TILES 19
#define FT    (F_ * T_)   // 77100, output row stride in floats
#define V_SOUND 343.0f
#define TWO_PI 6.28318530717958647692f

__device__ __forceinline__ v8f wmma_tile_k12(
    const float* cosT, const float* sinT,
    const float* cosO, const float* sinO,
    int nA, int tB, int halfSel)
{
    v8f acc = {};
    #pragma unroll
    for (int kk = 0; kk < 12; kk += 4) {
        // K channels: k = 2*p + {0:cos, 1:sin}. Lanes 0-15 carry K=kk,kk+1
        // (pair kk/2); lanes 16-31 carry K=kk+2,kk+3 (pair kk/2+1).
        const int p = (kk >> 1) + halfSel;
        v2f afrag, bfrag;
        afrag.x = cosT[p * NPAD + nA];
        afrag.y = sinT[p * NPAD + nA];
        bfrag.x = cosO[p * TPAD + tB];
        bfrag.y = sinO[p * TPAD + tB];
        acc = __builtin_amdgcn_wmma_f32_16x16x4_f32(
            /*neg_a=*/false, afrag, /*neg_b=*/false, bfrag,
            /*c_mod=*/(short)0, acc, /*reuse_a=*/false, /*reuse_b=*/false);
    }
    return acc;
}

__global__ __launch_bounds__(256) void CalculateDirectionFeature_kernel(
    const float* __restrict__ obs,    // (B*P, F, T)
    const float* __restrict__ azi,    // (B, N)
    const float* __restrict__ ele,    // (B, N)
    const float* __restrict__ pvec,   // (P, 3)
    const float* __restrict__ freq,   // (F,)
    float* __restrict__ out)          // (B, N, F, T)
{
    __shared__ float cosT[P_ * NPAD];   // cos(tpd) per pair, per n (zero padded)
    __shared__ float sinT[P_ * NPAD];
    __shared__ float cosO[P_ * TPAD];   // cos(obs) per pair, per t
    __shared__ float sinO[P_ * TPAD];

    const int bf  = blockIdx.x;
    const int b   = bf / F_;
    const int f   = bf % F_;
    const int tid = threadIdx.x;

    const float fr    = freq[f];
    const float scale = (TWO_PI / V_SOUND) * fr;   // tpd = scale * (pvec . r)

    // ---- Stage 1a: tpd trig table (6 pairs x 48 padded n) ----
    for (int i = tid; i < P_ * NPAD; i += 256) {
        const int p = i / NPAD;
        const int n = i % NPAD;
        float c = 0.0f, s = 0.0f;
        if (n < N_) {
            const float a = azi[b * N_ + n];
            const float e = ele[b * N_ + n];
            float sa, ca, se, ce;
            __sincosf(a, &sa, &ca);
            __sincosf(e, &se, &ce);
            const float rx = se * ca, ry = se * sa, rz = ce;
            const float dot = pvec[p * 3 + 0] * rx + pvec[p * 3 + 1] * ry +
                              pvec[p * 3 + 2] * rz;
            __sincosf(scale * dot, &s, &c);
        }
        cosT[i] = c;
        sinT[i] = s;
    }

    // ---- Stage 1b: observed-IPD trig table (6 pairs x 300 t, stride 304) ----
    for (int i = tid; i < P_ * TPAD; i += 256) {
        const int p = i / TPAD;
        const int t = i % TPAD;
        float c = 0.0f, s = 0.0f;
        if (t < T_) {
            const float o = obs[(((size_t)b * P_ + p) * F_ + f) * T_ + t];
            __sincosf(o, &s, &c);
        }
        cosO[i] = c;
        sinO[i] = s;
    }
    __syncthreads();

    // ---- Stage 2: 16x16 output tiles via V_WMMA_F32_16X16X4_F32, K=12 ----
    const int wave    = tid >> 5;     // 0..7
    const int lane    = tid & 31;
    const int halfSel = lane >> 4;    // 0: lanes 0-15, 1: lanes 16-31
    const int lane16  = lane & 15;

    // Uniform per-block output base for (b, f).
    float* const outBF = out + ((size_t)b * N_ * F_ + f) * T_;
    // Loop-invariant per-lane address term: C/D VGPR r of lanes 16-31 holds
    // row n0+8+r, lanes 0-15 hold row n0+r; column = t0 + lane16.
    const int laneOff = halfSel * (8 * FT) + lane16;

    // ---- Loop 1: fully in-bounds tiles (nt = 0,1), branch-free body ----
    for (int tile = wave; tile < 2 * TT_TILES; tile += 8) {
        const int nt = tile / TT_TILES;
        const int tt = tile % TT_TILES;
        const int n0 = nt * 16;
        // Last t-tile overlaps (t0 = 284): fully in-bounds, idempotent
        // duplicate writes on cols 284-287 (~1.3% extra store traffic).
        const int t0 = (tt == TT_TILES - 1) ? (T_ - 16) : tt * 16;

        v8f acc = wmma_tile_k12(cosT, sinT, cosO, sinO,
                                n0 + lane16, t0 + lane16, halfSel);

        float* op = outBF + ((size_t)(n0 * FT + t0) + laneOff);
        #pragma unroll
        for (int r = 0; r < 8; ++r) {
            op[(size_t)r * FT] = acc[r];   // constant IOFFSET r*308400
        }
    }

    // ---- Loop 2: n-tail tiles (nt = 2, rows 32..47; only 32..35 real) ----
    for (int tt = wave; tt < TT_TILES; tt += 8) {
        const int n0 = 32;
        const int t0 = (tt == TT_TILES - 1) ? (T_ - 16) : tt * 16;

        v8f acc = wmma_tile_k12(cosT, sinT, cosO, sinO,
                                n0 + lane16, t0 + lane16, halfSel);

        float* op = outBF + ((size_t)(n0 * FT + t0) + laneOff);
        if (halfSel == 0) {
            // Rows 32..35 live in VGPRs 0..3 of lanes 0-15 only: one
            // exec-mask set covers all four stores.
            #pragma unroll
            for (int r = 0; r < 4; ++r) {
                op[(size_t)r * FT] = acc[r];
            }
        }
    }
}

extern "C" void kernel_launch(void* const* d_in, const int* in_sizes, int n_in,
                              void* d_out, int out_size, void* d_ws, size_t ws_size,
                              hipStream_t stream) {
    (void)in_sizes; (void)n_in; (void)d_ws; (void)ws_size; (void)out_size;
    const float* obs  = (const float*)d_in[0];
    const float* azi  = (const float*)d_in[1];
    const float* ele  = (const float*)d_in[2];
    const float* pvec = (const float*)d_in[3];
    const float* freq = (const float*)d_in[4];
    float* out = (float*)d_out;

    dim3 grid(B_ * F_);   // 1028 blocks: one per (b, f)
    dim3 block(256);      // 8 wave32
    CalculateDirectionFeature_kernel<<<grid, block, 0, stream>>>(
        obs, azi, ele, pvec, freq, out);
}